// GNN_Layer_26491358281842
// MI455X (gfx1250) — compile-verified
//
#include <hip/hip_runtime.h>
#include <hip/hip_bf16.h>
#include <math.h>

#define N_NODES 50000
#define N_EDGES 800000
#define IN_DIM  128
#define HEADS   2
#define HEAD_DIM 64
#define OUT_DIM 128   // HEADS * HEAD_DIM
#define EPS_F   1e-16f

typedef __attribute__((ext_vector_type(2))) float v2f;
typedef __attribute__((ext_vector_type(8))) float v8f;

// ---- order-preserving uint encoding of float (for atomicMax over floats) ----
__device__ __forceinline__ unsigned int float_to_ordered(float f) {
    unsigned int u = __float_as_uint(f);
    return (u & 0x80000000u) ? ~u : (u | 0x80000000u);
}
__device__ __forceinline__ float ordered_to_float(unsigned int u) {
    u = (u & 0x80000000u) ? (u & 0x7fffffffu) : ~u;
    return __uint_as_float(u);
}

// ---------------------------------------------------------------------------
// Init: zero agg accumulator (d_out) + denom, set amax = encode(-inf)
// ---------------------------------------------------------------------------
__global__ __launch_bounds__(256)
void gnn_init(float* __restrict__ agg, unsigned int* __restrict__ amax,
              float* __restrict__ denom) {
    int tid = blockIdx.x * 256 + threadIdx.x;
    if (tid < N_NODES * OUT_DIM) agg[tid] = 0.0f;
    if (tid < N_NODES * HEADS) {
        amax[tid]  = float_to_ordered(-INFINITY);
        denom[tid] = 0.0f;
    }
}

// ---------------------------------------------------------------------------
// Projections: q/k/v/s = x @ W + b via V_WMMA_F32_16X16X4_F32 (full fp32).
// One wave per 16x16 output tile. grid = (N/16, OUT_DIM/16, 4 matrices).
// A (16x4 f32): lanes 0-15 hold M=0..15; lane-half picks K pair {0,1}/{2,3}.
// B (4x16 f32): lanes 0-15 hold N=0..15; lane-half picks K pair likewise.
// C/D (16x16 f32): acc[i] -> row (i + 8*laneHalf), col = lane&15.
// ---------------------------------------------------------------------------
__global__ __launch_bounds__(32)
void gnn_proj_wmma(const float* __restrict__ x,
                   const float* __restrict__ W0, const float* __restrict__ B0,
                   const float* __restrict__ W1, const float* __restrict__ B1,
                   const float* __restrict__ W2, const float* __restrict__ B2,
                   const float* __restrict__ W3, const float* __restrict__ B3,
                   float* __restrict__ q, float* __restrict__ k,
                   float* __restrict__ v, float* __restrict__ s) {
    const int lane = threadIdx.x & 31;
    const int row0 = blockIdx.x * 16;
    const int col0 = blockIdx.y * 16;
    const int mat  = blockIdx.z;

    const float* W; const float* bias; float* out;
    switch (mat) {
        case 0:  W = W0; bias = B0; out = q; break;
        case 1:  W = W1; bias = B1; out = k; break;
        case 2:  W = W2; bias = B2; out = v; break;
        default: W = W3; bias = B3; out = s; break;
    }

    const int m  = lane & 15;   // M for A, N for B/D
    const int kh = lane >> 4;   // which K pair this lane-half holds

    v8f acc = {};
    #pragma unroll
    for (int k0 = 0; k0 < IN_DIM; k0 += 4) {
        v2f a, b;
        a.x = x[(row0 + m) * IN_DIM + k0 + 2 * kh + 0];
        a.y = x[(row0 + m) * IN_DIM + k0 + 2 * kh + 1];
        b.x = W[(k0 + 2 * kh + 0) * OUT_DIM + col0 + m];
        b.y = W[(k0 + 2 * kh + 1) * OUT_DIM + col0 + m];
        acc = __builtin_amdgcn_wmma_f32_16x16x4_f32(
            /*neg_a=*/false, a, /*neg_b=*/false, b,
            /*c_mod=*/(short)0, acc, /*reuse_a=*/false, /*reuse_b=*/false);
    }

    const float bval = bias[col0 + m];
    #pragma unroll
    for (int i = 0; i < 8; ++i) {
        const int row = row0 + i + kh * 8;
        out[row * OUT_DIM + col0 + m] = acc[i] + bval;
    }
}

// ---------------------------------------------------------------------------
// Edge logits: alpha[e,h] = dot(q[dst,h,:], k[src,h,:]) / sqrt(D)
// One wave per edge; lane owns float4 of the 128-wide feature; 16-lane halves
// are the two heads; xor-shuffle reduce within each half.
// Also segment-max into amax[dst,h] via ordered-uint atomicMax.
// ---------------------------------------------------------------------------
__global__ __launch_bounds__(256)
void gnn_edge_logits(const int* __restrict__ ei,
                     const float* __restrict__ q, const float* __restrict__ k,
                     float* __restrict__ alpha, unsigned int* __restrict__ amax) {
    const int lane = threadIdx.x & 31;
    const int wave = threadIdx.x >> 5;
    const int e = blockIdx.x * 8 + wave;
    if (e >= N_EDGES) return;
    const int src = ei[e];
    const int dst = ei[N_EDGES + e];

    const float4 qv = *(const float4*)(q + (size_t)dst * OUT_DIM + lane * 4);
    const float4 kv = *(const float4*)(k + (size_t)src * OUT_DIM + lane * 4);
    float p = qv.x * kv.x + qv.y * kv.y + qv.z * kv.z + qv.w * kv.w;

    // reduce within each 16-lane half (xor offsets < 16 stay in the half)
    p += __shfl_xor(p, 8);
    p += __shfl_xor(p, 4);
    p += __shfl_xor(p, 2);
    p += __shfl_xor(p, 1);

    if ((lane & 15) == 0) {
        const int h = lane >> 4;
        const float a = p * 0.125f;  // 1/sqrt(64)
        alpha[(size_t)e * HEADS + h] = a;
        atomicMax(&amax[dst * HEADS + h], float_to_ordered(a));
    }
}

// ---------------------------------------------------------------------------
// exp(alpha - amax[dst]) in place; atomicAdd into denom[dst,h]
// ---------------------------------------------------------------------------
__global__ __launch_bounds__(256)
void gnn_edge_exp(const int* __restrict__ ei, float* __restrict__ alpha,
                  const unsigned int* __restrict__ amax,
                  float* __restrict__ denom) {
    const int tid = blockIdx.x * 256 + threadIdx.x;
    if (tid >= N_EDGES * HEADS) return;
    const int e = tid >> 1;
    const int h = tid & 1;
    const int dst = ei[N_EDGES + e];
    const float mx = ordered_to_float(amax[dst * HEADS + h]);
    const float ea = __expf(alpha[tid] - mx);
    alpha[tid] = ea;  // now holds ealpha
    atomicAdd(&denom[dst * HEADS + h], ea);
}

// ---------------------------------------------------------------------------
// Weighted scatter-aggregate: agg[dst,h,:] += (ealpha/denom) * v[src,h,:]
// One wave per edge, 4 f32 atomics per lane (global_atomic_add_f32, L2).
// ---------------------------------------------------------------------------
__global__ __launch_bounds__(256)
void gnn_aggregate(const int* __restrict__ ei, const float* __restrict__ vv,
                   const float* __restrict__ ealpha,
                   const float* __restrict__ denom, float* __restrict__ agg) {
    const int lane = threadIdx.x & 31;
    const int wave = threadIdx.x >> 5;
    const int e = blockIdx.x * 8 + wave;
    if (e >= N_EDGES) return;
    const int src = ei[e];
    const int dst = ei[N_EDGES + e];
    const int h = lane >> 4;

    const float w = ealpha[(size_t)e * HEADS + h] / (denom[dst * HEADS + h] + EPS_F);
    const float4 v4 = *(const float4*)(vv + (size_t)src * OUT_DIM + lane * 4);
    float* o = agg + (size_t)dst * OUT_DIM + lane * 4;
    atomicAdd(o + 0, w * v4.x);
    atomicAdd(o + 1, w * v4.y);
    atomicAdd(o + 2, w * v4.z);
    atomicAdd(o + 3, w * v4.w);
}

// ---------------------------------------------------------------------------
// out = relu(agg + skip)
// ---------------------------------------------------------------------------
__global__ __launch_bounds__(256)
void gnn_finalize(float* __restrict__ out, const float* __restrict__ s) {
    const int tid = blockIdx.x * 256 + threadIdx.x;
    if (tid < N_NODES * OUT_DIM) {
        const float r = out[tid] + s[tid];
        out[tid] = r > 0.0f ? r : 0.0f;
    }
}

// ---------------------------------------------------------------------------
extern "C" void kernel_launch(void* const* d_in, const int* in_sizes, int n_in,
                              void* d_out, int out_size, void* d_ws, size_t ws_size,
                              hipStream_t stream) {
    (void)in_sizes; (void)n_in; (void)out_size; (void)ws_size;

    const float* x  = (const float*)d_in[0];
    const int*   ei = (const int*)d_in[1];
    const float* Wq = (const float*)d_in[2];
    const float* bq = (const float*)d_in[3];
    const float* Wk = (const float*)d_in[4];
    const float* bk = (const float*)d_in[5];
    const float* Wv = (const float*)d_in[6];
    const float* bv = (const float*)d_in[7];
    const float* Ws = (const float*)d_in[8];
    const float* bs = (const float*)d_in[9];
    float* out = (float*)d_out;

    // workspace carve-up (256B aligned)
    char* ws = (char*)d_ws;
    size_t off = 0;
    auto alloc = [&](size_t bytes) -> void* {
        void* p = ws + off;
        off = (off + bytes + 255) & ~(size_t)255;
        return p;
    };
    float*        q      = (float*)alloc((size_t)N_NODES * OUT_DIM * sizeof(float));
    float*        k      = (float*)alloc((size_t)N_NODES * OUT_DIM * sizeof(float));
    float*        v      = (float*)alloc((size_t)N_NODES * OUT_DIM * sizeof(float));
    float*        s      = (float*)alloc((size_t)N_NODES * OUT_DIM * sizeof(float));
    float*        alpha  = (float*)alloc((size_t)N_EDGES * HEADS * sizeof(float));
    unsigned int* amax   = (unsigned int*)alloc((size_t)N_NODES * HEADS * sizeof(unsigned int));
    float*        denom  = (float*)alloc((size_t)N_NODES * HEADS * sizeof(float));

    // 1) init accumulators (d_out doubles as agg buffer)
    gnn_init<<<(N_NODES * OUT_DIM + 255) / 256, 256, 0, stream>>>(out, amax, denom);

    // 2) four projections via f32 WMMA: grid (N/16, 128/16, 4)
    gnn_proj_wmma<<<dim3(N_NODES / 16, OUT_DIM / 16, 4), 32, 0, stream>>>(
        x, Wq, bq, Wk, bk, Wv, bv, Ws, bs, q, k, v, s);

    // 3) per-edge logits + segment max
    gnn_edge_logits<<<(N_EDGES + 7) / 8, 256, 0, stream>>>(ei, q, k, alpha, amax);

    // 4) exp + segment sum
    gnn_edge_exp<<<(N_EDGES * HEADS + 255) / 256, 256, 0, stream>>>(ei, alpha, amax, denom);

    // 5) weighted scatter-aggregate into d_out
    gnn_aggregate<<<(N_EDGES + 7) / 8, 256, 0, stream>>>(ei, v, alpha, denom, out);

    // 6) skip + relu
    gnn_finalize<<<(N_NODES * OUT_DIM + 255) / 256, 256, 0, stream>>>(out, s);
}